// UserEmbeddings1_38465727103681
// MI455X (gfx1250) — compile-verified
//
#include <hip/hip_runtime.h>

#define NUM_USERS 100000
#define NUM_POIS  50000
#define NN        (NUM_USERS + NUM_POIS)   // 150000
#define DD        128
#define POI_DIM   320
#define EE        2000000
#define BB        1024
#define NEG_SLOPE 0.2f

typedef __attribute__((ext_vector_type(2))) float v2f;
typedef __attribute__((ext_vector_type(8))) float v8f;

// ---------------------------------------------------------------------------
// WMMA fp32 GEMM: C[rows x 128] = A[rows x K] * W[K x 128] (+ bias)
// 256 threads = 8 waves. Each wave computes a 16-row strip x 128 cols
// (8 output tiles of 16x16, v8f accumulators). W staged via LDS in 64-row
// chunks using a k-pair interleaved layout so each B fragment is a single
// aligned ds_load_b64 (pairs get merged into ds_load_2addr_b64):
//     lw[(k>>1)*256 + n*2 + (k&1)] = W[kc+k][n]
//
// Fragment layouts per CDNA5 ISA (32-bit 16x16x4):
//   A (16x4):  lane l, vgpr v -> A[M = l&15][K = 2*(l>>4) + v]
//   B (4x16):  lane l, vgpr v -> B[K = 2*(l>>4) + v][N = l&15]
//   C/D:       lane l, vgpr v -> C[M = v + 8*(l>>4)][N = l&15]
// ---------------------------------------------------------------------------
__global__ __launch_bounds__(256) void gemm_wmma_f32(
    const float* __restrict__ A, const float* __restrict__ W,
    const float* __restrict__ bias, float* __restrict__ C,
    int rows, int K)
{
    __shared__ float lw[64 * 128];   // 32 KB chunk of W (interleaved layout)
    __shared__ float lb[128];

    const int tid  = threadIdx.x;
    const int wave = tid >> 5;
    const int lane = tid & 31;
    const int hi   = lane >> 4;      // 0 or 1 (K-half select)
    const int nl   = lane & 15;      // M (for A) / N (for B,C,D)
    const int row0 = (blockIdx.x * 8 + wave) * 16;

    if (bias != nullptr && tid < 128) lb[tid] = bias[tid];

    v8f acc[8];
    #pragma unroll
    for (int t = 0; t < 8; ++t)
        #pragma unroll
        for (int i = 0; i < 8; ++i) acc[t][i] = 0.0f;

    // clamp A row so out-of-range waves still issue safe loads
    int ar = row0 + nl;
    if (ar >= rows) ar = rows - 1;

    for (int kc = 0; kc < K; kc += 64) {
        __syncthreads();
        // cooperative load of W[kc..kc+63][0..127] (8192 floats) into the
        // interleaved LDS layout (scatter as 4 b32 stores per float4)
        const float* wp = W + (size_t)kc * 128;
        for (int i = tid * 4; i < 64 * 128; i += 256 * 4) {
            float4 wv = *(const float4*)(&wp[i]);
            const int k = i >> 7, n = i & 127;
            float* dp = &lw[(k >> 1) * 256 + (k & 1) + n * 2];
            dp[0] = wv.x; dp[2] = wv.y; dp[4] = wv.z; dp[6] = wv.w;
        }
        __syncthreads();

        const float* ap = A + (size_t)ar * K + kc + 2 * hi;
        __builtin_prefetch(ap + 64, 0, 1);   // speculative: next K chunk of A
        for (int k4 = 0; k4 < 64; k4 += 4) {
            float2 av = *(const float2*)(ap + k4);
            v2f a; a.x = av.x; a.y = av.y;
            // this wave's k-pair index within the chunk: (k4 + 2*hi)/2
            const float* bpp = &lw[((k4 >> 1) + hi) * 256 + nl * 2];
            // materialize all 8 B fragments first so the DS loads issue as a
            // batch and the 8 WMMAs run back-to-back behind one wait
            v2f b[8];
            #pragma unroll
            for (int nt = 0; nt < 8; ++nt)
                b[nt] = *(const v2f*)(bpp + nt * 32);
            #pragma unroll
            for (int nt = 0; nt < 8; ++nt)
                acc[nt] = __builtin_amdgcn_wmma_f32_16x16x4_f32(
                    false, a, false, b[nt], (short)0, acc[nt], false, false);
        }
    }

    // store D (guard rows); add bias here if provided
    #pragma unroll
    for (int nt = 0; nt < 8; ++nt) {
        const int col = nt * 16 + nl;
        const float badd = (bias != nullptr) ? lb[col] : 0.0f;
        #pragma unroll
        for (int v = 0; v < 8; ++v) {
            const int r = row0 + v + 8 * hi;
            if (r < rows) C[(size_t)r * 128 + col] = acc[nt][v] + badd;
        }
    }
}

// ---------------------------------------------------------------------------
// degree / dinv
// ---------------------------------------------------------------------------
__global__ void init_deg_kernel(float* deg) {
    int i = blockIdx.x * blockDim.x + threadIdx.x;
    if (i < NN) deg[i] = 1.0f;                    // self-loop weight
}
__global__ void deg_accum_kernel(const int* __restrict__ dst, float* deg) {
    int e = blockIdx.x * blockDim.x + threadIdx.x;
    if (e < EE) atomicAdd(&deg[dst[e]], 1.0f);
}
__global__ void deg_to_dinv_kernel(float* deg) {
    int i = blockIdx.x * blockDim.x + threadIdx.x;
    if (i < NN) deg[i] = rsqrtf(deg[i]);
}

// ---------------------------------------------------------------------------
// agg init: agg = dinv^2 * h + b   (self-loop term + bias), float4 per thread
// ---------------------------------------------------------------------------
__global__ void init_agg_kernel(const float* __restrict__ h,
                                const float* __restrict__ dinv,
                                const float* __restrict__ b,
                                float* __restrict__ agg) {
    int idx = blockIdx.x * blockDim.x + threadIdx.x;     // over NN*32 float4s
    if (idx >= NN * (DD / 4)) return;
    int i  = idx >> 5;            // node
    int j4 = (idx & 31) * 4;      // starting column
    float dv = dinv[i]; dv *= dv;
    float4 hv = *(const float4*)(h + (size_t)i * DD + j4);
    float4 o;
    o.x = dv * hv.x + b[j4 + 0];
    o.y = dv * hv.y + b[j4 + 1];
    o.z = dv * hv.z + b[j4 + 2];
    o.w = dv * hv.w + b[j4 + 3];
    *(float4*)(agg + (size_t)i * DD + j4) = o;
}

// ---------------------------------------------------------------------------
// edge scatter: one wave32 per edge; lane handles 4 columns
// agg[dst] += h[src] * dinv[src] * dinv[dst]
// h/agg (76.8 MB each) both fit in the 192 MB L2 -> atomics stay on-chip
// ---------------------------------------------------------------------------
__global__ __launch_bounds__(256) void edge_scatter_kernel(
    const int* __restrict__ src, const int* __restrict__ dst,
    const float* __restrict__ dinv, const float* __restrict__ h,
    float* __restrict__ agg)
{
    int e = blockIdx.x * 8 + (threadIdx.x >> 5);
    if (e >= EE) return;
    const int lane = threadIdx.x & 31;
    const int s = src[e];
    const int d = dst[e];
    const float norm = dinv[s] * dinv[d];
    float4 hv = *(const float4*)(h + (size_t)s * DD + lane * 4);
    float* ag = agg + (size_t)d * DD + lane * 4;
    atomicAdd(ag + 0, hv.x * norm);
    atomicAdd(ag + 1, hv.y * norm);
    atomicAdd(ag + 2, hv.z * norm);
    atomicAdd(ag + 3, hv.w * norm);
}

// ---------------------------------------------------------------------------
// leaky relu in place
// ---------------------------------------------------------------------------
__global__ void leaky_kernel(float* x, int n) {
    int i = blockIdx.x * blockDim.x + threadIdx.x;
    if (i < n) {
        float v = x[i];
        x[i] = (v >= 0.0f) ? v : NEG_SLOPE * v;
    }
}

// ---------------------------------------------------------------------------
// batch gather: g[b] = x[user_idx[b]] + user_table[user_idx[b]]
// ---------------------------------------------------------------------------
__global__ void gather_add_kernel(const int* __restrict__ uidx,
                                  const float* __restrict__ x,
                                  const float* __restrict__ ut,
                                  float* __restrict__ g) {
    int idx = blockIdx.x * blockDim.x + threadIdx.x;
    if (idx >= BB * DD) return;
    int b = idx >> 7, j = idx & 127;
    int u = uidx[b];
    g[idx] = x[(size_t)u * DD + j] + ut[(size_t)u * DD + j];
}

// ---------------------------------------------------------------------------
extern "C" void kernel_launch(void* const* d_in, const int* in_sizes, int n_in,
                              void* d_out, int out_size, void* d_ws, size_t ws_size,
                              hipStream_t stream) {
    const int*   user_idx   = (const int*)  d_in[0];
    const float* poi_emb    = (const float*)d_in[1];
    const int*   edge_index = (const int*)  d_in[2];   // [2,E]: src then dst
    const float* user_table = (const float*)d_in[3];
    const float* Wp = (const float*)d_in[4];
    const float* bp = (const float*)d_in[5];
    const float* W1 = (const float*)d_in[6];
    const float* b1 = (const float*)d_in[7];
    const float* W2 = (const float*)d_in[8];
    const float* b2 = (const float*)d_in[9];
    const float* Wf = (const float*)d_in[10];
    const float* bf = (const float*)d_in[11];
    float* out = (float*)d_out;

    const int* src = edge_index;
    const int* dst = edge_index + EE;

    // workspace carve-out
    char* ws = (char*)d_ws;
    size_t off = 0;
    auto carve = [&](size_t bytes) -> void* {
        void* p = ws + off;
        off += (bytes + 255) & ~(size_t)255;
        return p;
    };
    float* dinv = (float*)carve((size_t)NN * 4);
    float* bufA = (float*)carve((size_t)NN * DD * 4);  // x0 -> agg2/x2
    float* bufB = (float*)carve((size_t)NN * DD * 4);  // h1 -> h2
    float* bufC = (float*)carve((size_t)NN * DD * 4);  // agg1/x1
    float* gbuf = (float*)carve((size_t)BB * DD * 4);

    const int T = 256;

    // x0[0:NUM_USERS] = user_table
    hipMemcpyAsync(bufA, user_table, (size_t)NUM_USERS * DD * 4,
                   hipMemcpyDeviceToDevice, stream);

    // x0[NUM_USERS:] = poi_emb @ Wp + bp
    gemm_wmma_f32<<<(NUM_POIS + 127) / 128, T, 0, stream>>>(
        poi_emb, Wp, bp, bufA + (size_t)NUM_USERS * DD, NUM_POIS, POI_DIM);

    // dinv = rsqrt(1 + indegree)
    init_deg_kernel<<<(NN + T - 1) / T, T, 0, stream>>>(dinv);
    deg_accum_kernel<<<(EE + T - 1) / T, T, 0, stream>>>(dst, dinv);
    deg_to_dinv_kernel<<<(NN + T - 1) / T, T, 0, stream>>>(dinv);

    const int gemmBlocks = (NN + 127) / 128;
    const int aggBlocks  = (NN * (DD / 4) + T - 1) / T;
    const int edgeBlocks = (EE + 7) / 8;
    const int eltBlocks  = (NN * DD + T - 1) / T;

    // ---- GCN layer 1 ----
    gemm_wmma_f32<<<gemmBlocks, T, 0, stream>>>(bufA, W1, nullptr, bufB, NN, DD);
    init_agg_kernel<<<aggBlocks, T, 0, stream>>>(bufB, dinv, b1, bufC);
    edge_scatter_kernel<<<edgeBlocks, T, 0, stream>>>(src, dst, dinv, bufB, bufC);
    leaky_kernel<<<eltBlocks, T, 0, stream>>>(bufC, NN * DD);

    // ---- GCN layer 2 ----
    gemm_wmma_f32<<<gemmBlocks, T, 0, stream>>>(bufC, W2, nullptr, bufB, NN, DD);
    init_agg_kernel<<<aggBlocks, T, 0, stream>>>(bufB, dinv, b2, bufA);
    edge_scatter_kernel<<<edgeBlocks, T, 0, stream>>>(src, dst, dinv, bufB, bufA);
    leaky_kernel<<<eltBlocks, T, 0, stream>>>(bufA, NN * DD);

    // ---- head: out = (x2[uidx] + user_table[uidx]) @ Wf + bf ----
    gather_add_kernel<<<(BB * DD + T - 1) / T, T, 0, stream>>>(
        user_idx, bufA, user_table, gbuf);
    gemm_wmma_f32<<<(BB + 127) / 128, T, 0, stream>>>(gbuf, Wf, bf, out, BB, DD);
}